// LeastSquareProjector_64665027609053
// MI455X (gfx1250) — compile-verified
//
#include <hip/hip_runtime.h>
#include <hip/hip_bf16.h>

// LeastSquareProjector for MI455X (gfx1250).
// x: (64, 256, 516) f32.  B = x[..., :512], Y = x[..., 512:516].
// G = B^T B (512x512), A = G + 0.01 I, W = A^-1 (B^T Y), out = W^T flat.
// Heavy math (Gram + Cholesky trailing update) on V_WMMA_F32_16X16X4_F32
// (full fp32 matrix pipe: solve condition number ~1e5 forbids reduced
// precision). LDS staging of small blocks uses the CDNA5 async-to-LDS path.

typedef float v2f __attribute__((ext_vector_type(2)));
typedef float v8f __attribute__((ext_vector_type(8)));

#define NBATCH 64
#define NN     256     // rows of B
#define LL     512     // cols of B == linear system size
#define DD     4       // RHS count
#define LDX    516     // row stride of x
#define NB     16      // Cholesky block size
#define KB     (LL/NB) // 32 block-columns

// ---- CDNA5 async global->LDS copy (ASYNCcnt path, ISA ch.10.7/15.18) ------
__device__ __forceinline__ void async_dword_to_lds(const float* g, float* l) {
    unsigned ldsOff = (unsigned)(uintptr_t)l;   // flat LDS addr: low 32 bits = LDS offset
    asm volatile("global_load_async_to_lds_b32 %0, %1, off"
                 :: "v"(ldsOff), "v"(g) : "memory");
}
__device__ __forceinline__ void wait_async0() {
    asm volatile("s_wait_asynccnt 0x0" ::: "memory");
}

// ---------------------------------------------------------------------------
// Kernel 1: G = B^T B + 0.01 I, lower triangle only.
// One wave per 64x32 macro tile (4x2 register-blocked 16x16 WMMA tiles):
// 12 dword loads per 8 v_wmma_f32_16x16x4_f32 (A-frags reused across the 2
// column tiles, B-frags across the 4 row tiles).
// A-matrix (16x4 f32) lane layout: lanes 0-15 -> M=lane, VGPR{0,1}=K{0,1};
// lanes 16-31 -> M=lane-16, VGPR{0,1}=K{2,3}. B-matrix (4x16) mirrored.
// C/D 16x16 f32: VGPR r -> M=r (lanes 0-15) / M=r+8 (lanes 16-31), N=lane&15.
// ---------------------------------------------------------------------------
__global__ __launch_bounds__(32)
void gram_kernel(const float* __restrict__ x, float* __restrict__ A, int baseBatch) {
    const int mc = blockIdx.x;                 // macro col (32 wide)
    const int mr = blockIdx.y;                 // macro row (64 tall)
    if (mc * 32 > mr * 64 + 63) return;        // macro tile entirely above diagonal
    const int b = blockIdx.z;

    const float* xb = x + (size_t)(baseBatch + b) * NN * LDX;
    float* Ab = A + (size_t)b * LL * LL;

    const int lane = threadIdx.x;
    const int half = lane >> 4;                // 0 -> K 0..1 ; 1 -> K 2..3
    const int li   = lane & 15;
    const int l0 = mr * 64, m0 = mc * 32;

    v8f c[4][2] = {};
    for (int n0 = 0; n0 < NN; n0 += 4) {
        const int k = n0 + half * 2;
        const float* r0 = xb + (size_t)k * LDX;
        const float* r1 = r0 + LDX;
        v2f a[4], bm[2];
#pragma unroll
        for (int rt = 0; rt < 4; ++rt) {
            a[rt].x = r0[l0 + rt * 16 + li];   // A[M, K=k]   = B[k][row]
            a[rt].y = r1[l0 + rt * 16 + li];   // A[M, K=k+1]
        }
#pragma unroll
        for (int ct = 0; ct < 2; ++ct) {
            bm[ct].x = r0[m0 + ct * 16 + li];  // B[K=k, N]   = B[k][col]
            bm[ct].y = r1[m0 + ct * 16 + li];
        }
#pragma unroll
        for (int rt = 0; rt < 4; ++rt)
#pragma unroll
            for (int ct = 0; ct < 2; ++ct)
                c[rt][ct] = __builtin_amdgcn_wmma_f32_16x16x4_f32(
                    false, a[rt], false, bm[ct], (short)0, c[rt][ct], false, false);
    }

#pragma unroll
    for (int rt = 0; rt < 4; ++rt) {
        const int row0 = l0 + rt * 16;
#pragma unroll
        for (int ct = 0; ct < 2; ++ct) {
            const int col0 = m0 + ct * 16;
            if (col0 > row0 + 15) continue;    // subtile above diagonal: unused
            const bool diag = (row0 == col0);  // wave-uniform
#pragma unroll
            for (int r = 0; r < 8; ++r) {
                float v = c[rt][ct][r];
                if (diag && (r + half * 8 == li)) v += 0.01f;   // + REG * I
                Ab[(size_t)(row0 + r + half * 8) * LL + col0 + li] = v;
            }
        }
    }
}

// ---------------------------------------------------------------------------
// Kernel 2: RHS = B^T Y  (512 x 4 per batch). Pure VALU, tiny.
// ---------------------------------------------------------------------------
__global__ __launch_bounds__(512)
void bty_kernel(const float* __restrict__ x, float* __restrict__ rhs, int baseBatch) {
    const int b = blockIdx.x;
    const float* xb = x + (size_t)(baseBatch + b) * NN * LDX;
    float* r = rhs + (size_t)b * LL * DD;
    const int l = threadIdx.x;
    float a0 = 0.f, a1 = 0.f, a2 = 0.f, a3 = 0.f;
    for (int n = 0; n < NN; ++n) {
        const float* row = xb + (size_t)n * LDX;
        float bb = row[l];
        a0 += bb * row[LL + 0];
        a1 += bb * row[LL + 1];
        a2 += bb * row[LL + 2];
        a3 += bb * row[LL + 3];
    }
    r[l * DD + 0] = a0; r[l * DD + 1] = a1; r[l * DD + 2] = a2; r[l * DD + 3] = a3;
}

// ---------------------------------------------------------------------------
// Kernel 3: blocked Cholesky (lower) of A in-place + blocked triangular
// solves for 4 RHS + transposed output store. One 256-thread WG per batch.
// Trailing update C -= P_bi * P_bj^T runs on WMMA f32 (A-fragment negated,
// since f32 NEG bits only cover the C operand). Diagonal blocks staged into
// LDS with GLOBAL_LOAD_ASYNC_TO_LDS_B32 + s_wait_asynccnt.
// ---------------------------------------------------------------------------
__global__ __launch_bounds__(256)
void cholsolve_kernel(float* __restrict__ A, const float* __restrict__ rhs,
                      float* __restrict__ out, int baseBatch) {
    __shared__ float P[LL][NB];        // 32 KB: current panel (tri-solved)
    __shared__ float Dg[NB][NB + 1];   // diag block (padded vs bank conflicts)
    __shared__ float Yl[LL][DD];       // 8 KB: RHS / solution

    const int b = blockIdx.x;
    float* Ab = A + (size_t)b * LL * LL;
    const float* rb = rhs + (size_t)b * LL * DD;

    const int tid  = threadIdx.x;
    const int lane = tid & 31;
    const int wave = tid >> 5;
    const int half = lane >> 4;
    const int li   = lane & 15;
    const int dr   = tid >> 4;         // 256 threads cover the 16x16 diag block
    const int dc   = tid & 15;

    // ---------------- factorization: A = L L^T (lower) ----------------
    for (int k = 0; k < KB; ++k) {
        const int j0 = k * NB;

        // stage diagonal block into LDS (async global->LDS path)
        async_dword_to_lds(&Ab[(size_t)(j0 + dr) * LL + j0 + dc], &Dg[dr][dc]);
        wait_async0();
        __syncthreads();

        // factor 16x16 diag block serially (~1.4 KFLOP; negligible)
        if (tid == 0) {
            for (int c = 0; c < NB; ++c) {
                float s = sqrtf(Dg[c][c]);
                Dg[c][c] = s;
                float inv = 1.0f / s;
                for (int r = c + 1; r < NB; ++r) Dg[r][c] *= inv;
                for (int c2 = c + 1; c2 < NB; ++c2) {
                    float f = Dg[c2][c];
                    for (int r = c2; r < NB; ++r) Dg[r][c2] -= Dg[r][c] * f;
                }
            }
        }
        __syncthreads();

        // write L_kk back (lower incl. diag)
        if (dc <= dr) Ab[(size_t)(j0 + dr) * LL + j0 + dc] = Dg[dr][dc];

        // panel solve: rows below diag block, P_row = A_row * L_kk^{-T}
        for (int row = j0 + NB + tid; row < LL; row += 256) {
            float p[NB];
#pragma unroll
            for (int cc = 0; cc < NB; ++cc) p[cc] = Ab[(size_t)row * LL + j0 + cc];
#pragma unroll
            for (int cc = 0; cc < NB; ++cc) {
                float v = p[cc];
                for (int t = 0; t < cc; ++t) v -= p[t] * Dg[cc][t];
                p[cc] = v / Dg[cc][cc];
            }
#pragma unroll
            for (int cc = 0; cc < NB; ++cc) {
                Ab[(size_t)row * LL + j0 + cc] = p[cc];
                P[row][cc] = p[cc];
            }
        }
        __syncthreads();

        // trailing update (the O(L^3) term) on WMMA f32:
        // for lower-triangle tiles (bi >= bj > k): C_tile -= P_bi * P_bj^T
        const int T = KB - 1 - k;
        const int ntiles = T * (T + 1) / 2;
        for (int t = wave; t < ntiles; t += 8) {
            int bi_rel = 0, acc = 0;
            while (acc + bi_rel + 1 <= t) { acc += bi_rel + 1; ++bi_rel; }
            int bj_rel = t - acc;
            int bi = k + 1 + bi_rel;
            int bj = k + 1 + bj_rel;

            v8f c;
#pragma unroll
            for (int r = 0; r < 8; ++r)
                c[r] = Ab[(size_t)(bi * NB + r + half * 8) * LL + bj * NB + li];

#pragma unroll
            for (int c0 = 0; c0 < NB; c0 += 4) {
                int kk = c0 + half * 2;
                v2f a, bm;
                a.x  = -P[bi * NB + li][kk];       // negate A -> C -= A*B
                a.y  = -P[bi * NB + li][kk + 1];
                bm.x =  P[bj * NB + li][kk];       // B[K][N] = P_bj[N][K]
                bm.y =  P[bj * NB + li][kk + 1];
                c = __builtin_amdgcn_wmma_f32_16x16x4_f32(
                        false, a, false, bm, (short)0, c, false, false);
            }
#pragma unroll
            for (int r = 0; r < 8; ++r)
                Ab[(size_t)(bi * NB + r + half * 8) * LL + bj * NB + li] = c[r];
        }
        __syncthreads();
    }

    // ---------------- blocked solves: L y = rhs ; L^T w = y ----------------
    for (int i = tid; i < LL * DD; i += 256) Yl[i >> 2][i & 3] = rb[i];
    __syncthreads();

    // forward substitution, 16-wide blocks (64 barrier rounds instead of 1024)
    for (int kb = 0; kb < KB; ++kb) {
        const int j0 = kb * NB;
        async_dword_to_lds(&Ab[(size_t)(j0 + dr) * LL + j0 + dc], &Dg[dr][dc]);
        wait_async0();
        __syncthreads();
        if (tid < DD) {                 // tiny 16-eq triangular solve per RHS
            for (int c = 0; c < NB; ++c) {
                float v = Yl[j0 + c][tid];
                for (int t = 0; t < c; ++t) v -= Dg[c][t] * Yl[j0 + t][tid];
                Yl[j0 + c][tid] = v / Dg[c][c];
            }
        }
        __syncthreads();
        for (int e = tid; e < (LL - j0 - NB) * DD; e += 256) {
            int i = j0 + NB + (e >> 2), d = e & 3;
            const float* Lrow = Ab + (size_t)i * LL + j0;
            float v = Yl[i][d];
#pragma unroll
            for (int c = 0; c < NB; ++c) v -= Lrow[c] * Yl[j0 + c][d];
            Yl[i][d] = v;
        }
        __syncthreads();
    }
    // backward substitution with L^T, 16-wide blocks
    for (int kb = KB - 1; kb >= 0; --kb) {
        const int j0 = kb * NB;
        async_dword_to_lds(&Ab[(size_t)(j0 + dr) * LL + j0 + dc], &Dg[dr][dc]);
        wait_async0();
        __syncthreads();
        if (tid < DD) {
            for (int c = NB - 1; c >= 0; --c) {
                float v = Yl[j0 + c][tid];
                for (int t = c + 1; t < NB; ++t) v -= Dg[t][c] * Yl[j0 + t][tid];
                Yl[j0 + c][tid] = v / Dg[c][c];
            }
        }
        __syncthreads();
        // rows above: y[i] -= sum_c L[j0+c][i] * w[j0+c]  (reads L rows: coalesced)
        for (int e = tid; e < j0 * DD; e += 256) {
            int i = e >> 2, d = e & 3;
            float v = Yl[i][d];
#pragma unroll
            for (int c = 0; c < NB; ++c) v -= Ab[(size_t)(j0 + c) * LL + i] * Yl[j0 + c][d];
            Yl[i][d] = v;
        }
        __syncthreads();
    }

    // out[b][d*512 + l] = W[l][d]   (the W^T reshape)
    float* ob = out + (size_t)(baseBatch + b) * (DD * LL);
    for (int i = tid; i < LL * DD; i += 256) {
        int l = i >> 2, d = i & 3;
        ob[d * LL + l] = Yl[l][d];
    }
}

// ---------------------------------------------------------------------------
// Host launcher. Chunks batches by available workspace; chunks serialize on
// `stream`, so this is graph-capture safe and deterministic.
// ---------------------------------------------------------------------------
extern "C" void kernel_launch(void* const* d_in, const int* in_sizes, int n_in,
                              void* d_out, int out_size, void* d_ws, size_t ws_size,
                              hipStream_t stream) {
    (void)in_sizes; (void)n_in; (void)out_size;
    const float* x = (const float*)d_in[0];
    float* out = (float*)d_out;

    const size_t perA = (size_t)LL * LL;          // floats per batch: A matrix
    const size_t perR = (size_t)LL * DD;          // floats per batch: RHS
    const size_t perBatchBytes = (perA + perR) * sizeof(float);

    int CB = (int)(ws_size / perBatchBytes);
    if (CB < 1) CB = 1;
    if (CB > NBATCH) CB = NBATCH;

    float* Aws = (float*)d_ws;
    float* Rws = Aws + (size_t)CB * perA;

    for (int base = 0; base < NBATCH; base += CB) {
        int nb = NBATCH - base;
        if (nb > CB) nb = CB;
        dim3 g1(LL / 32, LL / 64, nb);            // 16 x 8 macro tiles (64x32)
        gram_kernel<<<g1, 32, 0, stream>>>(x, Aws, base);
        bty_kernel<<<nb, LL, 0, stream>>>(x, Rws, base);
        cholsolve_kernel<<<nb, 256, 0, stream>>>(Aws, Rws, out, base);
    }
}